// GCNLayer_31817117729542
// MI455X (gfx1250) — compile-verified
//
#include <hip/hip_runtime.h>
#include <hip/hip_bf16.h>

typedef __attribute__((ext_vector_type(2))) float v2f;
typedef __attribute__((ext_vector_type(8))) float v8f;

#define CHUNK 512           // edges per wave
#define WAVES_PER_BLOCK 8   // 256 threads = 8 wave32

__global__ void gcn_zero_kernel(float* __restrict__ out, int n4) {
    int i = blockIdx.x * blockDim.x + threadIdx.x;
    int stride = gridDim.x * blockDim.x;
    float4 z = make_float4(0.f, 0.f, 0.f, 0.f);
    for (; i < n4; i += stride)
        reinterpret_cast<float4*>(out)[i] = z;
}

__global__ void __launch_bounds__(256)
gcn_spmm_wmma_kernel(const int* __restrict__ row,
                     const int* __restrict__ col,
                     const float* __restrict__ vals,
                     const float* __restrict__ embeds,
                     float* __restrict__ out,
                     int E, int n_nodes) {
    const int lane = threadIdx.x & 31;
    const int m    = lane & 15;   // feature index (N) == window-row index (M) for A
    const int hi   = lane >> 4;   // wave half: selects K pair {0,1} or {2,3}

    // wave id, forced scalar so edge-stream loads scalarize
    const int w = __builtin_amdgcn_readfirstlane(
        (int)(blockIdx.x * WAVES_PER_BLOCK + (threadIdx.x >> 5)));
    long e0 = (long)w * CHUNK;
    if (e0 >= E) return;
    long e_end = e0 + CHUNK;
    if (e_end > E) e_end = E;

    v8f acc = {0.f, 0.f, 0.f, 0.f, 0.f, 0.f, 0.f, 0.f};
    int base_row = row[e0];   // uniform

    long e = e0;
    for (; e + 3 < e_end; e += 4) {
        if (((e - e0) & 63) == 0) {
            // linear edge streams: pull next 1KB toward L2/L0
            __builtin_prefetch(row  + e + 256, 0, 1);
            __builtin_prefetch(col  + e + 256, 0, 1);
            __builtin_prefetch(vals + e + 256, 0, 1);
        }
        int4   rq = *reinterpret_cast<const int4*>(row + e);
        int4   cq = *reinterpret_cast<const int4*>(col + e);
        float4 vq = *reinterpret_cast<const float4*>(vals + e);

        if (rq.w >= base_row + 16) {
            // slide window: flush 16x16 accumulator tile with f32 atomics
            #pragma unroll
            for (int i = 0; i < 8; ++i) {
                int rr = base_row + i + hi * 8;   // C/D layout: VGPR i -> M=i (+8 hi half)
                float v = acc[i];
                if (rr < n_nodes && v != 0.0f)
                    unsafeAtomicAdd(out + (long)rr * 16 + m, v);
            }
            acc = (v8f){0.f, 0.f, 0.f, 0.f, 0.f, 0.f, 0.f, 0.f};
            base_row = rq.x;
        }

        // per-lane K pair: half 0 -> edges e+0,e+1 ; half 1 -> edges e+2,e+3
        int   rk0 = hi ? rq.z : rq.x;
        int   rk1 = hi ? rq.w : rq.y;
        int   ck0 = hi ? cq.z : cq.x;
        int   ck1 = hi ? cq.w : cq.y;
        float vk0 = hi ? vq.z : vq.x;
        float vk1 = hi ? vq.w : vq.y;

        // gather embeds rows (B fragment): 64B contiguous per half-wave, L2-resident
        float b0 = embeds[(long)ck0 * 16 + m];
        float b1 = embeds[(long)ck1 * 16 + m];

        if (rq.w < base_row + 16) {
            // A[m,k] = one-hot(row) * val ; acc(16x16) += A(16x4) @ B(4x16)
            v2f a, b;
            a.x = (rk0 - base_row == m) ? vk0 : 0.0f;
            a.y = (rk1 - base_row == m) ? vk1 : 0.0f;
            b.x = b0;
            b.y = b1;
            acc = __builtin_amdgcn_wmma_f32_16x16x4_f32(
                false, a, false, b, (short)0, acc, false, false);
        } else {
            // ultra-rare: 4 edges span >= 16 rows even after rebase -> direct atomics
            unsafeAtomicAdd(out + (long)rk0 * 16 + m, vk0 * b0);
            unsafeAtomicAdd(out + (long)rk1 * 16 + m, vk1 * b1);
        }
    }

    // final window flush
    #pragma unroll
    for (int i = 0; i < 8; ++i) {
        int rr = base_row + i + hi * 8;
        float v = acc[i];
        if (rr < n_nodes && v != 0.0f)
            unsafeAtomicAdd(out + (long)rr * 16 + m, v);
    }

    // tail edges (< 4): half-wave direct atomic path
    for (; e < e_end; ++e) {
        if (hi == 0) {
            int rr = row[e], cc = col[e];
            float vv = vals[e];
            unsafeAtomicAdd(out + (long)rr * 16 + m,
                            vv * embeds[(long)cc * 16 + m]);
        }
    }
}

extern "C" void kernel_launch(void* const* d_in, const int* in_sizes, int n_in,
                              void* d_out, int out_size, void* d_ws, size_t ws_size,
                              hipStream_t stream) {
    const int*   row    = (const int*)d_in[0];
    const int*   col    = (const int*)d_in[1];
    const float* vals   = (const float*)d_in[2];
    const float* embeds = (const float*)d_in[3];
    float*       out    = (float*)d_out;

    const int E       = in_sizes[0];
    const int n_nodes = in_sizes[3] / 16;   // D = 16

    // zero-init output (harness poisons d_out)
    int n4 = out_size / 4;
    int zb = (n4 + 255) / 256;
    if (zb > 4096) zb = 4096;
    gcn_zero_kernel<<<zb, 256, 0, stream>>>(out, n4);

    long n_waves = ((long)E + CHUNK - 1) / CHUNK;
    int  blocks  = (int)((n_waves + WAVES_PER_BLOCK - 1) / WAVES_PER_BLOCK);
    gcn_spmm_wmma_kernel<<<blocks, 32 * WAVES_PER_BLOCK, 0, stream>>>(
        row, col, vals, embeds, out, E, n_nodes);
}